// GCNBody_39376260169850
// MI455X (gfx1250) — compile-verified
//
#include <hip/hip_runtime.h>
#include <hip/hip_bf16.h>

#define NN 100000
#define NE 3200000
#define DH 32
#define BN_EPS 1e-5f

typedef __attribute__((ext_vector_type(2))) float v2f;
typedef __attribute__((ext_vector_type(8))) float v8f;

// ---------------------------------------------------------------- utilities
__global__ void fill_zero_kernel(float* __restrict__ p, int n) {
    int i = blockIdx.x * blockDim.x + threadIdx.x;
    if (i < n) p[i] = 0.0f;
}

// -------------------------------------------- encoder linear1 + BN statistics
__global__ __launch_bounds__(256)
void enc1_stats_kernel(const float* __restrict__ x, const float* __restrict__ w1,
                       const float* __restrict__ b1, float* __restrict__ h1,
                       float* __restrict__ gsum, float* __restrict__ gsq) {
    __shared__ float ssum[DH];
    __shared__ float ssq[DH];
    const int i = blockIdx.x * blockDim.x + threadIdx.x;
    if (threadIdx.x < DH) { ssum[threadIdx.x] = 0.0f; ssq[threadIdx.x] = 0.0f; }
    __syncthreads();
    if (i < NN) {
        const float x0 = x[2 * i], x1 = x[2 * i + 1];
#pragma unroll
        for (int c = 0; c < DH; ++c) {
            float h = fmaf(w1[2 * c], x0, fmaf(w1[2 * c + 1], x1, b1[c]));
            h1[(size_t)i * DH + c] = h;
            atomicAdd(&ssum[c], h);        // ds_add_f32
            atomicAdd(&ssq[c], h * h);
        }
    }
    __syncthreads();
    if (threadIdx.x < DH) {
        unsafeAtomicAdd(&gsum[threadIdx.x], ssum[threadIdx.x]);
        unsafeAtomicAdd(&gsq[threadIdx.x], ssq[threadIdx.x]);
    }
}

// ----------------------------------------------------------- BN finalization
__global__ void bn_finalize_kernel(const float* __restrict__ gsum, const float* __restrict__ gsq,
                                   const float* __restrict__ gamma, const float* __restrict__ beta,
                                   float* __restrict__ scale, float* __restrict__ shift) {
    const int c = threadIdx.x;
    if (c < DH) {
        const float inv_n = 1.0f / (float)NN;
        float mean = gsum[c] * inv_n;
        float var  = gsq[c] * inv_n - mean * mean;
        float s = gamma[c] * rsqrtf(var + BN_EPS);
        scale[c] = s;
        shift[c] = beta[c] - mean * s;
    }
}

// ------------------------ BN + PReLU + linear2 via V_WMMA_F32_16X16X4_F32 ---
// One wave per 16-node tile. A: 16x4 f32 fragments (lanes 0-15 hold K=2h,2h+1),
// B: w2^T 4x16 fragments, C/D: 16x16 f32 (8 VGPRs). 16 WMMAs per wave.
__global__ __launch_bounds__(256)
void enc2_wmma_kernel(const float* __restrict__ h1, const float* __restrict__ scale,
                      const float* __restrict__ shift, const float* __restrict__ prelu_a,
                      const float* __restrict__ w2, const float* __restrict__ b2,
                      float* __restrict__ henc) {
    const int wave_id = (blockIdx.x * blockDim.x + threadIdx.x) >> 5;
    const int lane = threadIdx.x & 31;
    int tile = wave_id;
    const int n_tiles = NN / 16;                 // 6250, N divisible by 16
    if (tile >= n_tiles) tile = n_tiles - 1;     // clamp: keep EXEC all-ones for WMMA
    const int half = lane >> 4;                  // 0: lanes 0-15, 1: lanes 16-31
    const int m = lane & 15;
    const int row = tile * 16 + m;
    const float alpha = prelu_a[0];

    // A fragments: BN + PReLU fused in
    v2f a[8];
#pragma unroll
    for (int kb = 0; kb < 8; ++kb) {
        const int k0 = 4 * kb + 2 * half;
        float v0 = fmaf(h1[(size_t)row * DH + k0],     scale[k0],     shift[k0]);
        float v1 = fmaf(h1[(size_t)row * DH + k0 + 1], scale[k0 + 1], shift[k0 + 1]);
        v0 = (v0 >= 0.0f) ? v0 : alpha * v0;
        v1 = (v1 >= 0.0f) ? v1 : alpha * v1;
        a[kb].x = v0;
        a[kb].y = v1;
    }

    v8f c0 = {};
    v8f c1 = {};
#pragma unroll
    for (int kb = 0; kb < 8; ++kb) {
        const int k0 = 4 * kb + 2 * half;
        v2f b0, b1v;
        b0.x  = w2[(size_t)m * DH + k0];            // B[k][n] = w2[n][k], n-tile 0
        b0.y  = w2[(size_t)m * DH + k0 + 1];
        b1v.x = w2[(size_t)(16 + m) * DH + k0];     // n-tile 1
        b1v.y = w2[(size_t)(16 + m) * DH + k0 + 1];
        c0 = __builtin_amdgcn_wmma_f32_16x16x4_f32(false, a[kb], false, b0,
                                                   (short)0, c0, false, false);
        c1 = __builtin_amdgcn_wmma_f32_16x16x4_f32(false, a[kb], false, b1v,
                                                   (short)0, c1, false, false);
    }

    const float bias0 = b2[m], bias1 = b2[16 + m];
#pragma unroll
    for (int v = 0; v < 8; ++v) {
        const int r = tile * 16 + v + 8 * half;     // C layout: VGPR v -> M=v (+8 for hi lanes)
        henc[(size_t)r * DH + m]      = c0[v] + bias0;
        henc[(size_t)r * DH + 16 + m] = c1[v] + bias1;
    }
}

// -------------------------------------------------------------------- degree
__global__ __launch_bounds__(256)
void degree_kernel(const int* __restrict__ dst, float* __restrict__ deg) {
    int e = blockIdx.x * blockDim.x + threadIdx.x;
    if (e < NE) unsafeAtomicAdd(&deg[dst[e]], 1.0f);
}

__global__ __launch_bounds__(256)
void dinv_kernel(float* __restrict__ deg) {
    int i = blockIdx.x * blockDim.x + threadIdx.x;
    if (i < NN) deg[i] = rsqrtf(deg[i] + 1.0f);   // + self loop
}

// ------------------- fold gcn_w / gcn_b / wb / bb into one 32-vector + const
__global__ void head_fold_kernel(const float* __restrict__ gcn_w, const float* __restrict__ gcn_b,
                                 const float* __restrict__ wb, const float* __restrict__ bb,
                                 float* __restrict__ vfold) {
    const int k = threadIdx.x;
    if (k < DH) {
        float acc = 0.0f;
#pragma unroll
        for (int n = 0; n < DH; ++n) acc = fmaf(wb[n], gcn_w[(size_t)n * DH + k], acc);
        vfold[k] = acc;
    }
    if (k == 0) {
        float acc = bb[0];
#pragma unroll
        for (int n = 0; n < DH; ++n) acc = fmaf(wb[n], gcn_b[n], acc);
        vfold[DH] = acc;
    }
}

// ------------------------------------- edge gather / scale / scatter-add ----
// 8 threads per edge, 4 channels each: float4 gather + 4 f32 atomics.
__global__ __launch_bounds__(256)
void scatter_kernel(const int* __restrict__ src, const int* __restrict__ dst,
                    const float* __restrict__ dinv, const float* __restrict__ henc,
                    float* __restrict__ agg) {
    const long long tid = (long long)blockIdx.x * blockDim.x + threadIdx.x;
    const int e = (int)(tid >> 3);
    const int j = (int)(tid & 7);
    if (e >= NE) return;
    const int s = src[e], d = dst[e];
    const float w = dinv[s] * dinv[d];
    const float4 hv = ((const float4*)(henc + (size_t)s * DH))[j];
    float* ap = agg + (size_t)d * DH + 4 * j;
    unsafeAtomicAdd(ap + 0, hv.x * w);
    unsafeAtomicAdd(ap + 1, hv.y * w);
    unsafeAtomicAdd(ap + 2, hv.z * w);
    unsafeAtomicAdd(ap + 3, hv.w * w);
}

// ------------------- self-loop + folded GCN-linear + score head, fused ------
__global__ __launch_bounds__(256)
void score_kernel(const float* __restrict__ agg, const float* __restrict__ henc,
                  const float* __restrict__ dinv, const float* __restrict__ vfold,
                  float* __restrict__ out) {
    const int i = blockIdx.x * blockDim.x + threadIdx.x;
    if (i >= NN) return;
    const float di = dinv[i];
    const float dii = di * di;
    float acc = vfold[DH];
    const float4* ap = (const float4*)(agg + (size_t)i * DH);
    const float4* hp = (const float4*)(henc + (size_t)i * DH);
    const float4* vp = (const float4*)vfold;
#pragma unroll
    for (int j = 0; j < 8; ++j) {
        const float4 av = ap[j];
        const float4 hv = hp[j];
        const float4 vv = vp[j];
        acc = fmaf(fmaf(hv.x, dii, av.x), vv.x, acc);
        acc = fmaf(fmaf(hv.y, dii, av.y), vv.y, acc);
        acc = fmaf(fmaf(hv.z, dii, av.z), vv.z, acc);
        acc = fmaf(fmaf(hv.w, dii, av.w), vv.w, acc);
    }
    out[i] = acc;
}

// ---------------------------------------------------------------------------
extern "C" void kernel_launch(void* const* d_in, const int* in_sizes, int n_in,
                              void* d_out, int out_size, void* d_ws, size_t ws_size,
                              hipStream_t stream) {
    const float* x       = (const float*)d_in[0];
    const int*   eidx    = (const int*)d_in[1];    // [2, NE]: src then dst
    const float* w1      = (const float*)d_in[2];
    const float* b1      = (const float*)d_in[3];
    const float* gamma   = (const float*)d_in[4];
    const float* beta    = (const float*)d_in[5];
    const float* prelu_a = (const float*)d_in[6];
    const float* w2      = (const float*)d_in[7];
    const float* b2      = (const float*)d_in[8];
    const float* gcn_w   = (const float*)d_in[9];
    const float* gcn_b   = (const float*)d_in[10];
    const float* wb      = (const float*)d_in[11];
    const float* bb      = (const float*)d_in[12];
    float* out = (float*)d_out;

    const int* e_src = eidx;
    const int* e_dst = eidx + NE;

    // Workspace layout (floats): h1/agg [NN*32] | henc [NN*32] | deg/dinv [NN]
    //                            | stats [128] (sum,sq,scale,shift) | vfold [33]
    float* ws    = (float*)d_ws;
    float* h1    = ws;                              // reused as agg after enc2
    float* henc  = ws + (size_t)NN * DH;
    float* deg   = henc + (size_t)NN * DH;          // becomes dinv in-place
    float* stats = deg + NN;
    float* vfold = stats + 128;

    const int B = 256;

    // 1) zero accumulators
    fill_zero_kernel<<<1, 128, 0, stream>>>(stats, 128);
    fill_zero_kernel<<<(NN + B - 1) / B, B, 0, stream>>>(deg, NN);

    // 2) encoder linear1 + BN stats
    enc1_stats_kernel<<<(NN + B - 1) / B, B, 0, stream>>>(x, w1, b1, h1,
                                                          stats, stats + 32);
    // 3) BN scale/shift
    bn_finalize_kernel<<<1, 32, 0, stream>>>(stats, stats + 32, gamma, beta,
                                             stats + 64, stats + 96);
    // 4) BN + PReLU + linear2 (WMMA), 8 waves/block, 1 tile/wave
    {
        const int n_tiles = NN / 16;                       // 6250
        const int blocks = (n_tiles + 7) / 8;              // 782
        enc2_wmma_kernel<<<blocks, B, 0, stream>>>(h1, stats + 64, stats + 96,
                                                   prelu_a, w2, b2, henc);
    }
    // 5) zero agg (reuses h1 region; enc2 has consumed h1)
    fill_zero_kernel<<<((NN * DH) + B - 1) / B, B, 0, stream>>>(h1, NN * DH);

    // 6) degrees + normalization
    degree_kernel<<<(NE + B - 1) / B, B, 0, stream>>>(e_dst, deg);
    dinv_kernel<<<(NN + B - 1) / B, B, 0, stream>>>(deg);

    // 7) fold gcn linear + score head into one vector
    head_fold_kernel<<<1, 32, 0, stream>>>(gcn_w, gcn_b, wb, bb, vfold);

    // 8) edge gather/scatter (8 threads per edge)
    {
        const long long threads = (long long)NE * 8;
        const int blocks = (int)((threads + B - 1) / B);   // 100000
        scatter_kernel<<<blocks, B, 0, stream>>>(e_src, e_dst, deg, henc, h1);
    }
    // 9) self-loop + fused final linear/head
    score_kernel<<<(NN + B - 1) / B, B, 0, stream>>>(h1, henc, deg, vfold, out);
}